// Get_Cat_Feat_Tgt_69672959475884
// MI455X (gfx1250) — compile-verified
//
#include <hip/hip_runtime.h>
#include <hip/hip_bf16.h>

typedef __attribute__((ext_vector_type(2))) float v2f;
typedef __attribute__((ext_vector_type(8))) float v8f;

#define K_NN   32
#define MTILE  16
#define NCHUNK 128          // 8 waves * 16 cols
#define N2TOT  32768
#define FDIM   32
#define MTOT   4096         // K_topk * C

__global__ __launch_bounds__(256) void knn_cat_feat_kernel(
    const float* __restrict__ cand,      // (4096, 3)  candidate points (flattened B,K,C,3)
    const float* __restrict__ tgt_xyz,   // (32768, 3)
    const float* __restrict__ tgt_feat,  // (32768, 32)
    float* __restrict__ out)             // (4096, 32, 35)
{
    __shared__ float qx[MTILE], qy[MTILE], qz[MTILE], qn2[MTILE];
    __shared__ unsigned long long pend[MTILE][NCHUNK];    // 16 KB survivor staging
    __shared__ unsigned long long best[MTILE][K_NN];      //  4 KB, sorted ascending
    __shared__ unsigned long long curMax[MTILE];          // running 32nd-smallest key
    __shared__ unsigned int pendCnt[MTILE];
    __shared__ float wbuf[MTILE][K_NN];

    const int tid   = threadIdx.x;
    const int lane  = tid & 31;
    const int wave  = tid >> 5;
    const int mbase = blockIdx.x * MTILE;

    // ---- init per-query state ----
    if (tid < MTILE) {
        int m = mbase + tid;
        float x = cand[m * 3 + 0], y = cand[m * 3 + 1], z = cand[m * 3 + 2];
        qx[tid] = x; qy[tid] = y; qz[tid] = z;
        qn2[tid] = x * x + y * y + z * z;
        curMax[tid]  = 0xFFFFFFFFFFFFFFFFull;
        pendCnt[tid] = 0u;
    }
    for (int i = tid; i < MTILE * K_NN; i += 256)
        ((unsigned long long*)best)[i] = 0xFFFFFFFFFFFFFFFFull;
    __syncthreads();

    // ---- A fragment: 16x4 f32, identical in all waves (K=3 zero-padded to 4) ----
    const int arow  = lane & 15;         // M row
    const int khalf = lane >> 4;         // 0 -> K{0,1}, 1 -> K{2,3}
    v2f afrag;
    afrag.x = (khalf == 0) ? qx[arow] : qz[arow];
    afrag.y = (khalf == 0) ? qy[arow] : 0.0f;

    // ---- per-lane register copies of |q|^2 for this lane's 8 C-rows ----
    float qn2r[8];
#pragma unroll
    for (int j = 0; j < 8; ++j) qn2r[j] = qn2[j + 8 * khalf];

    // ---- main loop over reference points: 256 iterations of 128 columns ----
    for (int nbase = 0; nbase < N2TOT; nbase += NCHUNK) {
        // Hoist all 8 thresholds into registers FIRST: curMax is stable for the
        // whole filter phase (only the insert phase after the barrier writes it),
        // so these DS loads overlap the global load + WMMA below instead of
        // serializing into the compare chain.
        unsigned long long thr[8];
#pragma unroll
        for (int j = 0; j < 8; ++j) thr[j] = curMax[j + 8 * khalf];

        const int col = nbase + wave * 16 + (lane & 15);   // this lane's N column
        const float rxv = tgt_xyz[col * 3 + 0];
        const float ryv = tgt_xyz[col * 3 + 1];
        const float rzv = tgt_xyz[col * 3 + 2];
        const float rn2 = rxv * rxv + ryv * ryv + rzv * rzv;

        v2f bfrag;                                          // B: 4x16 f32
        bfrag.x = (khalf == 0) ? rxv : rzv;
        bfrag.y = (khalf == 0) ? ryv : 0.0f;

        v8f acc = {};
        acc = __builtin_amdgcn_wmma_f32_16x16x4_f32(false, afrag, false, bfrag,
                                                    (short)0, acc, false, false);

        // ---- filter directly from accumulator registers ----
#pragma unroll
        for (int j = 0; j < 8; ++j) {
            const int mrow = j + 8 * khalf;                 // C/D layout row
            float d2 = fmaxf(qn2r[j] + rn2 - 2.0f * acc[j], 0.0f);
            unsigned long long key =
                ((unsigned long long)__float_as_uint(d2) << 32) | (unsigned int)col;
            if (key < thr[j]) {
                unsigned int p = atomicAdd(&pendCnt[mrow], 1u);
                pend[mrow][p] = key;
            }
        }
        __syncthreads();

        // ---- serial insertion of survivors (one thread per query) ----
        if (tid < MTILE) {
            const int qi = tid;
            const unsigned int cnt = pendCnt[qi];
            for (unsigned int p = 0; p < cnt; ++p) {
                unsigned long long key = pend[qi][p];
                if (key < best[qi][K_NN - 1]) {
                    int pos = K_NN - 1;
                    while (pos > 0 && best[qi][pos - 1] > key) {
                        best[qi][pos] = best[qi][pos - 1];
                        --pos;
                    }
                    best[qi][pos] = key;
                }
            }
            pendCnt[qi] = 0u;
            curMax[qi]  = best[qi][K_NN - 1];
        }
        __syncthreads();
    }

    // ---- weights: w = dist / sum(dist) ----
    if (tid < MTILE) {
        float s = 0.0f;
#pragma unroll
        for (int j = 0; j < K_NN; ++j) {
            float d2 = __uint_as_float((unsigned int)(best[tid][j] >> 32));
            float d  = sqrtf(d2);
            wbuf[tid][j] = d;
            s += d;
        }
        float inv = 1.0f / s;
#pragma unroll
        for (int j = 0; j < K_NN; ++j) wbuf[tid][j] *= inv;
    }
    __syncthreads();

    // ---- epilogue: one (query, neighbor) pair per unit of work ----
    for (int pair = tid; pair < MTILE * K_NN; pair += 256) {
        const int qi = pair >> 5;
        const int j  = pair & 31;
        const unsigned int idx = (unsigned int)(best[qi][j] & 0xFFFFFFFFu);
        const int m = mbase + qi;
        float* o = out + ((size_t)m * K_NN + j) * (3 + FDIM);
        o[0] = tgt_xyz[idx * 3 + 0] - qx[qi];
        o[1] = tgt_xyz[idx * 3 + 1] - qy[qi];
        o[2] = tgt_xyz[idx * 3 + 2] - qz[qi];
        // feature row is 128B-aligned: gather as float4, scale by w[f] (ref quirk:
        // weight indexed by FEATURE dim, broadcast over neighbor dim)
        const float4* fr4 = (const float4*)(tgt_feat + (size_t)idx * FDIM);
#pragma unroll
        for (int f4 = 0; f4 < FDIM / 4; ++f4) {
            float4 v = fr4[f4];
            o[3 + 4 * f4 + 0] = v.x * wbuf[qi][4 * f4 + 0];
            o[3 + 4 * f4 + 1] = v.y * wbuf[qi][4 * f4 + 1];
            o[3 + 4 * f4 + 2] = v.z * wbuf[qi][4 * f4 + 2];
            o[3 + 4 * f4 + 3] = v.w * wbuf[qi][4 * f4 + 3];
        }
    }
}

extern "C" void kernel_launch(void* const* d_in, const int* in_sizes, int n_in,
                              void* d_out, int out_size, void* d_ws, size_t ws_size,
                              hipStream_t stream) {
    (void)in_sizes; (void)n_in; (void)d_ws; (void)ws_size; (void)out_size;
    const float* cand     = (const float*)d_in[0];   // candidate_pts (1,512,8,3)
    const float* tgt_xyz  = (const float*)d_in[3];   // tgt_pts_xyz (1,32768,3)
    const float* tgt_feat = (const float*)d_in[4];   // tgt_deep_feat_pts (1,32768,32)
    float* out = (float*)d_out;

    dim3 grid(MTOT / MTILE);   // 256 blocks: one per 16-query tile
    dim3 block(256);           // 8 wave32s
    knn_cat_feat_kernel<<<grid, block, 0, stream>>>(cand, tgt_xyz, tgt_feat, out);
}